// DecoderRNN_41377714930190
// MI455X (gfx1250) — compile-verified
//
#include <hip/hip_runtime.h>
#include <math.h>

typedef __attribute__((ext_vector_type(2))) float v2f;
typedef __attribute__((ext_vector_type(8))) float v8f;

#define EMBED 256
#define HIDDEN 512
#define G4 (4 * HIDDEN)          // 2048 gate rows
#define VOCAB 32000
#define BB 16
#define TT 63
#define S (BB * (TT + 1))        // 1024 sequential steps

#define NWG 32                   // persistent workgroups for the scan
#define UNITS_PER_WG (HIDDEN / NWG)    // 16 hidden units per WG
#define ROWS_PER_WG (4 * UNITS_PER_WG) // 64 gate rows per WG
#define LTH 512                  // threads in scan WG (8 threads per gate row)

// ---- CDNA5 async memory->LDS helpers (ASYNCcnt path) -------------------
__device__ __forceinline__ unsigned lds_addr_of(const void* p) {
    // generic LDS pointer = {shared aperture, lds_offset[31:0]}
    return (unsigned)(unsigned long long)(uintptr_t)p;
}
__device__ __forceinline__ void async_load_b32(unsigned lds, const void* gaddr) {
    asm volatile("global_load_async_to_lds_b32 %0, %1, off"
                 :: "v"(lds), "v"((unsigned long long)(uintptr_t)gaddr)
                 : "memory");
}
__device__ __forceinline__ void async_load_b128(unsigned lds, const void* gaddr) {
    asm volatile("global_load_async_to_lds_b128 %0, %1, off"
                 :: "v"(lds), "v"((unsigned long long)(uintptr_t)gaddr)
                 : "memory");
}
__device__ __forceinline__ void wait_async0() {
    asm volatile("s_wait_asynccnt 0x0" ::: "memory");
}

// ---- gfx1250 hardware transcendental --------------------------------
__device__ __forceinline__ float tanh_hw(float x) {
    float y;
    // v_nop covers the TRANS-op result-use hazard (ISA 5.7.4)
    asm volatile("v_tanh_f32 %0, %1\n\tv_nop" : "=v"(y) : "v"(x));
    return y;
}
__device__ __forceinline__ float sigmoid_hw(float x) {
    return 0.5f * tanh_hw(0.5f * x) + 0.5f;
}

// ---------------------------------------------------------------- init
__global__ void init_kernel(float* __restrict__ hbuf, unsigned* __restrict__ cnt) {
    int t = blockIdx.x * blockDim.x + threadIdx.x;
    if (t < 2 * HIDDEN) hbuf[t] = 0.0f;
    if (t == 0) *cnt = 0u;
}

// ------------------------------------------------------- embed gather
__global__ void gather_kernel(const float* __restrict__ features,
                              const int* __restrict__ captions,
                              const float* __restrict__ emb,
                              float* __restrict__ xs) {
    int s = blockIdx.x;
    int b = s >> 6;
    int t = s & 63;
    const float* src = (t == 0)
        ? (features + (size_t)b * EMBED)
        : (emb + (size_t)captions[b * TT + (t - 1)] * EMBED);
    xs[(size_t)s * EMBED + threadIdx.x] = src[threadIdx.x];
}

// ------------------------------------------------ WMMA f32 GEMM, 4x4 blocked
// C[1024,N] = A[1024,K] @ B[N,K]^T + bias1 (+ bias2)
template <int K, int N>
__global__ void __launch_bounds__(256) gemm_wmma_kernel(
    const float* __restrict__ A, const float* __restrict__ B,
    const float* __restrict__ bias1, const float* __restrict__ bias2,
    float* __restrict__ C) {
    const int lane = threadIdx.x & 31;
    const int wave = threadIdx.x >> 5;
    const int tile = blockIdx.x * 8 + wave;   // 64x64 macro-tile id
    const int NT64 = N / 64;
    const int m0 = (tile / NT64) * 64;
    const int n0 = (tile % NT64) * 64;
    const int mi = lane & 15;
    const int kp = (lane >> 4) << 1;          // lanes16-31 hold K+2,K+3

    const float* ap[4];
    const float* bp[4];
#pragma unroll
    for (int i = 0; i < 4; ++i) {
        ap[i] = A + (size_t)(m0 + 16 * i + mi) * K + kp;
        bp[i] = B + (size_t)(n0 + 16 * i + mi) * K + kp;
    }

    v8f c[4][4] = {};
#pragma unroll 2
    for (int k0 = 0; k0 < K; k0 += 4) {
        v2f a[4], b[4];
#pragma unroll
        for (int i = 0; i < 4; ++i) {
            a[i] = v2f{ ap[i][k0], ap[i][k0 + 1] };
            b[i] = v2f{ bp[i][k0], bp[i][k0 + 1] };
        }
#pragma unroll
        for (int i = 0; i < 4; ++i)
#pragma unroll
            for (int j = 0; j < 4; ++j)
                c[i][j] = __builtin_amdgcn_wmma_f32_16x16x4_f32(
                    false, a[i], false, b[j], (short)0, c[i][j], false, false);
    }

#pragma unroll
    for (int j = 0; j < 4; ++j) {
        const int n = n0 + 16 * j + mi;
        float bv = bias1[n];
        if (bias2) bv += bias2[n];
#pragma unroll
        for (int i = 0; i < 4; ++i) {
            const int mbase = m0 + 16 * i + ((lane >> 4) << 3);
#pragma unroll
            for (int r = 0; r < 8; ++r)
                C[(size_t)(mbase + r) * N + n] = c[i][j][r] + bv;
        }
    }
}

// --------------------------------------------------------- LSTM scan
__global__ void __launch_bounds__(LTH, 1) lstm_scan_kernel(
    const float* __restrict__ W_hh,   // [2048,512]
    const float* __restrict__ xp,     // [1024,2048]
    float* __restrict__ hs,           // [1024,512]
    float* __restrict__ hbuf,         // [2][512] ping-pong
    unsigned* __restrict__ cnt) {
    extern __shared__ __attribute__((aligned(16))) float smem[];
    float* Wsh  = smem;                          // 64*512 floats = 128 KB
    float* hsh  = Wsh + ROWS_PER_WG * HIDDEN;    // 512
    float* gsh  = hsh + HIDDEN;                  // 64
    float* csh  = gsh + ROWS_PER_WG;             // 16
    float* xpsh = csh + UNITS_PER_WG;            // 64

    const int wg  = blockIdx.x;
    const int tid = threadIdx.x;

    // Stage this WG's 64 gate rows of W_hh into LDS via async b128 DMA.
    // local row r -> global row (r/16)*512 + wg*16 + (r%16)
    const int CHUNKS = ROWS_PER_WG * HIDDEN / 4;     // 16B chunks
    for (int cch = tid; cch < CHUNKS; cch += LTH) {
        int i = cch << 2;
        int r = i >> 9;
        int k = i & 511;
        int grow = (r >> 4) * HIDDEN + wg * UNITS_PER_WG + (r & 15);
        async_load_b128(lds_addr_of(&Wsh[i]),
                        W_hh + (size_t)grow * HIDDEN + k);
    }
    wait_async0();
    if (tid < UNITS_PER_WG) csh[tid] = 0.0f;
    __syncthreads();

    const int r   = tid >> 3;       // gate row 0..63
    const int seg = tid & 7;        // 64-wide K segment
    const float* wrow = &Wsh[r * HIDDEN + seg * 64];
    const unsigned la_h  = lds_addr_of(&hsh[tid]);
    const unsigned la_xp = (tid < ROWS_PER_WG) ? lds_addr_of(&xpsh[tid]) : 0u;
    const int grow_t = (tid >> 4) * HIDDEN + wg * UNITS_PER_WG + (tid & 15);

    for (int s = 0; s < S; ++s) {
        // async-fetch this step's h vector and xp gate-bias slice in parallel
        if (tid < ROWS_PER_WG)   // waves 0-1 (wave-uniform)
            async_load_b32(la_xp, xp + (size_t)s * G4 + grow_t);
        async_load_b32(la_h, hbuf + (s & 1) * HIDDEN + tid);
        wait_async0();
        __syncthreads();

        const float* hseg = &hsh[seg * 64];
        float p = 0.0f;
#pragma unroll
        for (int k = 0; k < 64; ++k) p = fmaf(wrow[k], hseg[k], p);
        p += __shfl_xor(p, 4, 32);
        p += __shfl_xor(p, 2, 32);
        p += __shfl_xor(p, 1, 32);
        if (seg == 0) gsh[r] = p + xpsh[r];
        __syncthreads();

        if (tid < UNITS_PER_WG) {
            float iv = sigmoid_hw(gsh[tid]);
            float fv = sigmoid_hw(gsh[UNITS_PER_WG + tid]);
            float gv = tanh_hw(gsh[2 * UNITS_PER_WG + tid]);
            float ov = sigmoid_hw(gsh[3 * UNITS_PER_WG + tid]);
            float cn = fv * csh[tid] + iv * gv;
            csh[tid] = cn;
            float hv = ov * tanh_hw(cn);
            int j = wg * UNITS_PER_WG + tid;
            hbuf[((s + 1) & 1) * HIDDEN + j] = hv;
            hs[(size_t)s * HIDDEN + j] = hv;
        }
        __threadfence();
        __syncthreads();
        if (tid == 0) {
            __hip_atomic_fetch_add(cnt, 1u, __ATOMIC_ACQ_REL,
                                   __HIP_MEMORY_SCOPE_AGENT);
            unsigned tgt = (unsigned)NWG * (unsigned)(s + 1);
            while (__hip_atomic_load(cnt, __ATOMIC_ACQUIRE,
                                     __HIP_MEMORY_SCOPE_AGENT) < tgt)
                __builtin_amdgcn_s_sleep(1);
        }
        __syncthreads();
    }
}

// ------------------------------------------------- softmax (in place)
// Whole 32000-float row staged in LDS via async b128 DMA (128 KB of the
// 320 KB WGP LDS): one global read pass, exp evaluated once, one write pass.
__global__ void __launch_bounds__(256) softmax_kernel(float* __restrict__ probs) {
    extern __shared__ __attribute__((aligned(16))) float row[];  // VOCAB floats
    __shared__ float red[8];
    float* p = probs + (size_t)blockIdx.x * VOCAB;
    const int tid = threadIdx.x;

    for (int i4 = tid; i4 < VOCAB / 4; i4 += 256)
        async_load_b128(lds_addr_of(&row[i4 << 2]), p + ((size_t)i4 << 2));
    wait_async0();
    __syncthreads();

    float m = -INFINITY;
    for (int i = tid; i < VOCAB; i += 256) m = fmaxf(m, row[i]);
#pragma unroll
    for (int o = 16; o > 0; o >>= 1) m = fmaxf(m, __shfl_xor(m, o, 32));
    if ((tid & 31) == 0) red[tid >> 5] = m;
    __syncthreads();
    float mm = red[0];
#pragma unroll
    for (int j = 1; j < 8; ++j) mm = fmaxf(mm, red[j]);
    __syncthreads();

    float sum = 0.0f;
    for (int i = tid; i < VOCAB; i += 256) {
        float e = expf(row[i] - mm);
        row[i] = e;                              // cache exp in LDS
        sum += e;
    }
#pragma unroll
    for (int o = 16; o > 0; o >>= 1) sum += __shfl_xor(sum, o, 32);
    if ((tid & 31) == 0) red[tid >> 5] = sum;
    __syncthreads();
    float tot = 0.0f;
#pragma unroll
    for (int j = 0; j < 8; ++j) tot += red[j];
    float inv = 1.0f / tot;
    __syncthreads();

    for (int i = tid; i < VOCAB; i += 256) p[i] = row[i] * inv;
}

// ---------------------------------------------------------------- host
extern "C" void kernel_launch(void* const* d_in, const int* in_sizes, int n_in,
                              void* d_out, int out_size, void* d_ws, size_t ws_size,
                              hipStream_t stream) {
    const float* features = (const float*)d_in[0];
    const int*   captions = (const int*)d_in[1];
    const float* emb      = (const float*)d_in[2];
    const float* W_ih     = (const float*)d_in[3];
    const float* W_hh     = (const float*)d_in[4];
    const float* b_ih     = (const float*)d_in[5];
    const float* b_hh     = (const float*)d_in[6];
    const float* W_out    = (const float*)d_in[7];
    const float* b_out    = (const float*)d_in[8];
    float* out = (float*)d_out;

    float* ws   = (float*)d_ws;
    float* xs   = ws;                        // 1024*256
    float* xp   = xs + (size_t)S * EMBED;    // 1024*2048
    float* hs   = xp + (size_t)S * G4;       // 1024*512
    float* hbuf = hs + (size_t)S * HIDDEN;   // 2*512
    unsigned* cnt = (unsigned*)(hbuf + 2 * HIDDEN);

    init_kernel<<<4, 256, 0, stream>>>(hbuf, cnt);
    gather_kernel<<<S, EMBED, 0, stream>>>(features, captions, emb, xs);

    gemm_wmma_kernel<EMBED, G4><<<(16 * (G4 / 64)) / 8, 256, 0, stream>>>(
        xs, W_ih, b_ih, b_hh, xp);

    size_t shbytes = (size_t)(ROWS_PER_WG * HIDDEN + HIDDEN + ROWS_PER_WG +
                              UNITS_PER_WG + ROWS_PER_WG) * sizeof(float);
    lstm_scan_kernel<<<NWG, LTH, shbytes, stream>>>(W_hh, xp, hs, hbuf, cnt);

    gemm_wmma_kernel<HIDDEN, VOCAB><<<(16 * (VOCAB / 64)) / 8, 256, 0, stream>>>(
        hs, W_out, b_out, (const float*)nullptr, out);

    softmax_kernel<<<S, 256, (size_t)VOCAB * sizeof(float), stream>>>(out);
}